// MultiModalFusion_62534723829955
// MI455X (gfx1250) — compile-verified
//
#include <hip/hip_runtime.h>

typedef __attribute__((ext_vector_type(2))) float v2f;
typedef __attribute__((ext_vector_type(8))) float v8f;

constexpr int Bn = 4, Cn = 128, Hn = 100, Wn = 256;
constexpr int Sn = Hn * Wn;       // 25600 spatial positions
constexpr int K2 = 2 * Cn;        // 256 (concat channels)
constexpr int TS = 16;            // spatial positions per workgroup
constexpr int ROWS = TS * Bn;     // 64 GEMM rows per workgroup
constexpr int RS = Cn + 4;        // padded LDS row stride (132 floats)

// LDS layout (float offsets)
constexpr int OFF_RAW = 0;                    // raw feats [2][ROWS][RS] (reused as out buffer)
constexpr int OFF_F   = OFF_RAW + 2*ROWS*RS;  // conv result f [ROWS][RS]
constexpr int OFF_R   = OFF_F   + ROWS*RS;    // retrieved    [ROWS][RS]
constexpr int OFF_AW  = OFF_R   + ROWS*RS;    // adapter [2][Cn]
constexpr int OFF_CB  = OFF_AW  + 2*Cn;       // conv_b [Cn]
constexpr int OFF_MB  = OFF_CB  + Cn;         // mlp_b [Cn]
constexpr int OFF_W4  = OFF_MB  + Cn;         // combined attn weights [TS][4][4]
constexpr int SMEM_FLOATS = OFF_W4 + TS*16;   // 34560 floats = 138240 B

#if __has_builtin(__builtin_amdgcn_global_load_async_to_lds_b32)
#define HAVE_ASYNC_LDS 1
#else
#define HAVE_ASYNC_LDS 0
#endif

typedef __attribute__((address_space(1))) int gint;   // global (AS1) int
typedef __attribute__((address_space(3))) int lint;   // LDS (AS3) int

// Pre-transpose weights into k-pair-interleaved B layout: Bp[(k>>1)][n][k&1].
// One b64 load then yields both B-fragment VGPRs for a lane's k-pair.
__global__ void prep_weights(const float* __restrict__ conv_w,  // [Cn][K2]
                             const float* __restrict__ mlp_w,   // [Cn][Cn]
                             float* __restrict__ B1,            // [K2/2][Cn][2]
                             float* __restrict__ B2) {          // [Cn/2][Cn][2]
  int i = blockIdx.x * blockDim.x + threadIdx.x;
  if (i < K2 * Cn) {
    int k = i >> 7, n = i & (Cn - 1);
    B1[((k >> 1) * Cn + n) * 2 + (k & 1)] = conv_w[n * K2 + k];
  }
  if (i < Cn * Cn) {
    int k = i >> 7, n = i & (Cn - 1);
    B2[((k >> 1) * Cn + n) * 2 + (k & 1)] = mlp_w[n * Cn + k];
  }
}

__global__ void __launch_bounds__(256)
fused_fusion_kernel(const float* __restrict__ feat0,
                    const float* __restrict__ feat1,
                    const float* __restrict__ adapter,  // [2][Cn]
                    const float* __restrict__ conv_b,   // [Cn]
                    const float* __restrict__ mlp_b,    // [Cn]
                    const float* __restrict__ B1,       // conv_w^T, k-pair interleaved
                    const float* __restrict__ B2,       // mlp_w^T,  k-pair interleaved
                    float* __restrict__ out)            // [Bn][Cn][Sn]
{
  extern __shared__ float sm[];
  const int t  = threadIdx.x;
  const int s0 = blockIdx.x * TS;

  float* rawL = sm + OFF_RAW;
  float* fL   = sm + OFF_F;
  float* rL   = sm + OFF_R;
  float* awL  = sm + OFF_AW;
  float* cbL  = sm + OFF_CB;
  float* mbL  = sm + OFF_MB;
  float* w4L  = sm + OFF_W4;

  // ---------------- stage inputs (coalesced along w) ----------------
  awL[t] = adapter[t];                       // 256 == 2*Cn
  if (t < Cn) { cbL[t] = conv_b[t]; mbL[t] = mlp_b[t]; }
  for (int e = t; e < 2 * Bn * Cn * TS; e += 256) {
    int sl = e & (TS - 1);
    int q  = e >> 4;
    int c  = q & (Cn - 1); q >>= 7;
    int b  = q & 3;
    int m  = q >> 2;
    const float* src = m ? feat1 : feat0;
    int gidx = (b * Cn + c) * Sn + s0 + sl;
    int lidx = m * ROWS * RS + (sl * Bn + b) * RS + c;
#if HAVE_ASYNC_LDS
    // CDNA5 async global->LDS copy (ASYNCcnt-tracked, no VGPR round trip)
    __builtin_amdgcn_global_load_async_to_lds_b32(
        (gint*)(src + gidx), (lint*)(rawL + lidx), 0, 0);
#else
    rawL[lidx] = src[gidx];
#endif
  }
#if HAVE_ASYNC_LDS
#if __has_builtin(__builtin_amdgcn_s_wait_asynccnt)
  __builtin_amdgcn_s_wait_asynccnt(0);
#else
  asm volatile("s_wait_asynccnt 0x0" ::: "memory");
#endif
#endif
  __syncthreads();

  const int wave = t >> 5;
  const int lane = t & 31;
  const int lh   = lane & 15;
  const int hi   = lane >> 4;
  const int n    = wave * 16 + lh;           // output column 0..127 (one col-tile per wave)

  // ------- conv GEMM: f = (adapter-scaled cat) @ conv_w^T + conv_b -------
  {
    v8f acc[4];
#pragma unroll
    for (int rt = 0; rt < 4; ++rt) acc[rt] = (v8f)0.0f;
    for (int k = 0; k < K2; k += 4) {
      int kk = k + hi * 2;                    // even
      v2f bf = *(const v2f*)(B1 + ((kk >> 1) * Cn + n) * 2);  // one b64 load
      int m   = k >> 7;                       // modality for this k-block
      int cb2 = (k & (Cn - 1)) + hi * 2;      // channel within modality
      float sa0 = awL[m * Cn + cb2];
      float sa1 = awL[m * Cn + cb2 + 1];
      const float* rbase = rawL + m * ROWS * RS + cb2;
#pragma unroll
      for (int rt = 0; rt < 4; ++rt) {
        const float* ap = rbase + (rt * 16 + lh) * RS;
        v2f af; af.x = ap[0] * sa0; af.y = ap[1] * sa1;
        acc[rt] = __builtin_amdgcn_wmma_f32_16x16x4_f32(
            false, af, false, bf, (short)0, acc[rt], false, false);
      }
    }
    float cbv = cbL[n];
#pragma unroll
    for (int rt = 0; rt < 4; ++rt)
#pragma unroll
      for (int v = 0; v < 8; ++v) {
        int row = rt * 16 + (hi ? v + 8 : v);
        fL[row * RS + n] = acc[rt][v] + cbv;
      }
  }
  __syncthreads();

  // ------- per-location cross-agent attention (one thread per (s,b,d)) -------
  {
    int s = t >> 4, b = (t >> 2) & 3, d = t & 3;
    const float* fd = fL + (s * Bn + d) * RS;
    const float* r0 = rawL + (s * Bn + b) * RS;
    const float* r1 = rawL + ROWS * RS + (s * Bn + b) * RS;
    float sc0 = 0.f, sc1 = 0.f;
    for (int c = 0; c < Cn; ++c) {
      float fv = fd[c];
      sc0 = fmaf(r0[c], fv, sc0);
      sc1 = fmaf(r1[c], fv, sc1);
    }
    const float inv = 0.08838834764831845f;   // 1/sqrt(128)
    sc0 *= inv; sc1 *= inv;
    float m0 = fmaxf(sc0, __shfl_xor(sc0, 1, 4)); m0 = fmaxf(m0, __shfl_xor(m0, 2, 4));
    float m1 = fmaxf(sc1, __shfl_xor(sc1, 1, 4)); m1 = fmaxf(m1, __shfl_xor(m1, 2, 4));
    float e0 = __expf(sc0 - m0), e1 = __expf(sc1 - m1);
    float t0 = e0 + __shfl_xor(e0, 1, 4); t0 += __shfl_xor(t0, 2, 4);
    float t1 = e1 + __shfl_xor(e1, 1, 4); t1 += __shfl_xor(t1, 2, 4);
    w4L[t] = e0 / t0 + e1 / t1;               // retrieved sums over m -> weights add
  }
  __syncthreads();

  // ------- retrieved = (attn0+attn1) @ f  (4x4 @ 4x128 per s) -------
  for (int idx = t; idx < ROWS * Cn; idx += 256) {
    int row = idx >> 7, c = idx & (Cn - 1);
    const float* w  = w4L + row * 4;
    const float* fb = fL + (row & ~3) * RS + c;   // rows s*4 .. s*4+3
    rL[row * RS + c] = w[0] * fb[0] + w[1] * fb[RS] + w[2] * fb[2 * RS] + w[3] * fb[3 * RS];
  }
  __syncthreads();

  // ------- MLP GEMM + relu + residual: out = f + relu(retrieved @ mlp_w^T + mlp_b) -------
  {
    v8f acc[4];
#pragma unroll
    for (int rt = 0; rt < 4; ++rt) acc[rt] = (v8f)0.0f;
    for (int k = 0; k < Cn; k += 4) {
      int kk = k + hi * 2;                    // even
      v2f bf = *(const v2f*)(B2 + ((kk >> 1) * Cn + n) * 2);  // one b64 load
#pragma unroll
      for (int rt = 0; rt < 4; ++rt) {
        const float* ap = rL + (rt * 16 + lh) * RS + kk;
        v2f af; af.x = ap[0]; af.y = ap[1];
        acc[rt] = __builtin_amdgcn_wmma_f32_16x16x4_f32(
            false, af, false, bf, (short)0, acc[rt], false, false);
      }
    }
    float mbv = mbL[n];
    float* oL = rawL;                           // raw feats no longer needed: reuse as out buf
#pragma unroll
    for (int rt = 0; rt < 4; ++rt)
#pragma unroll
      for (int v = 0; v < 8; ++v) {
        int row = rt * 16 + (hi ? v + 8 : v);
        float g = acc[rt][v] + mbv;
        oL[row * RS + n] = fL[row * RS + n] + fmaxf(g, 0.f);
      }
  }
  __syncthreads();

  // ------- coalesced transposed store: out[b][c][s0+sl] -------
  {
    int sl = t & 15;
    for (int p = t >> 4; p < Bn * Cn; p += 16) {
      int b = p >> 7, c = p & (Cn - 1);
      out[(b * Cn + c) * Sn + s0 + sl] = rawL[(sl * Bn + b) * RS + c];
    }
  }
}

extern "C" void kernel_launch(void* const* d_in, const int* in_sizes, int n_in,
                              void* d_out, int out_size, void* d_ws, size_t ws_size,
                              hipStream_t stream) {
  const float* feat0   = (const float*)d_in[0];
  const float* feat1   = (const float*)d_in[1];
  const float* adapter = (const float*)d_in[2];
  const float* conv_w  = (const float*)d_in[3];
  const float* conv_b  = (const float*)d_in[4];
  const float* mlp_w   = (const float*)d_in[5];
  const float* mlp_b   = (const float*)d_in[6];
  float* out = (float*)d_out;

  float* B1 = (float*)d_ws;          // [K2/2][Cn][2] = 32768 floats
  float* B2 = B1 + K2 * Cn;          // [Cn/2][Cn][2] = 16384 floats

  prep_weights<<<(K2 * Cn + 255) / 256, 256, 0, stream>>>(conv_w, mlp_w, B1, B2);

  size_t smem = SMEM_FLOATS * sizeof(float);  // ~135 KB of the 320 KB WGP LDS
  (void)hipFuncSetAttribute((const void*)fused_fusion_kernel,
                            hipFuncAttributeMaxDynamicSharedMemorySize, (int)smem);
  fused_fusion_kernel<<<Sn / TS, 256, smem, stream>>>(
      feat0, feat1, adapter, conv_b, mlp_b, B1, B2, out);
}